// TemporalHashEncoding_7902739825027
// MI455X (gfx1250) — compile-verified
//
#include <hip/hip_runtime.h>
#include <cmath>
#include <cstdint>
#include <cstddef>

// Instant-NGP-style 4D temporal hash encoding for MI455X (gfx1250).
// Pure gather/stream workload: no matrix math, so no WMMA — the chip-specific
// paths used are async global->LDS staging (ASYNCcnt), s_wait_asynccnt, and
// non-temporal b128 stores to preserve L2 (192 MB) for the hash tables
// (~180 MB total -> L2-resident gathers).

typedef float v4f __attribute__((ext_vector_type(4)));

struct Level {
  unsigned long long M;    // Lemire magic: UINT64_MAX / size + 1
  unsigned long long tbl;  // device pointer to table (size x 4 floats)
  float sxyz;              // spatial scale (sr)
  float st;                // temporal scale (tr)
  unsigned int size;       // hashmap size (from in_sizes -> exact)
  unsigned int _pad;
};

struct Params {
  Level lv[16];
};

__global__ __launch_bounds__(256) void temporal_hash_encode(
    const float* __restrict__ coords, float* __restrict__ out,
    Params p, unsigned int nPoints) {
  __shared__ v4f s_coord[16];  // 16 points per 256-thread block

  const unsigned int t = threadIdx.x;
  const unsigned int blockPoint = blockIdx.x * 16u;

  // Stage this block's 16 coordinate float4s into LDS via the CDNA5
  // asynchronous global->LDS DMA path (tracked by ASYNCcnt).
  if (t < 16u && (blockPoint + t) < nPoints) {
    unsigned long long gaddr =
        (unsigned long long)(const void*)(coords + (size_t)(blockPoint + t) * 4u);
    unsigned int ldsOff = (unsigned int)(size_t)(&s_coord[t]);  // low 32 bits of flat LDS addr = LDS offset
    asm volatile("global_load_async_to_lds_b128 %0, %1, off"
                 :: "v"(ldsOff), "v"(gaddr)
                 : "memory");
  }
  asm volatile("s_wait_asynccnt 0x0" ::: "memory");
  __syncthreads();

  const unsigned int tid = blockIdx.x * 256u + t;
  const unsigned int point = tid >> 4;
  if (point >= nPoints) return;
  const unsigned int level = tid & 15u;

  const v4f c = s_coord[t >> 4];

  const Level L = p.lv[level];

  // grid = floor(coord * scale) as int32 (coords in [0,1) -> non-negative)
  const int gx = (int)floorf(c.x * L.sxyz);
  const int gy = (int)floorf(c.y * L.sxyz);
  const int gz = (int)floorf(c.z * L.sxyz);
  const int gw = (int)floorf(c.w * L.st);

  // int32 wrapping multiplies + XOR combine (done in uint32 to avoid UB)
  const unsigned int h = (unsigned int)gx * 73856093u
                       ^ (unsigned int)gy * 19349663u
                       ^ (unsigned int)gz * 83492791u
                       ^ (unsigned int)gw * 50331653u;
  const int hs = (int)h;
  const unsigned int ua = (hs < 0) ? (0u - h) : h;  // |h| as uint (2^31 for INT_MIN)

  // idx = ua % size via Lemire fastmod (no integer division)
  const unsigned long long low = L.M * (unsigned long long)ua;
  unsigned int r = (unsigned int)__umul64hi(low, (unsigned long long)L.size);
  // numpy floored-mod semantics for abs(INT_MIN) == INT_MIN (wraps negative):
  // remainder(-2^31, size) = (size - (2^31 % size)) % size
  if (hs == (-2147483647 - 1)) r = r ? (L.size - r) : 0u;

  const v4f* tb = (const v4f*)L.tbl;
  const v4f f = tb[r];

  // out flat index: point*64 + level*4 == tid*4 -> perfectly coalesced b128.
  // Non-temporal: don't let the 256MB output stream evict L2-resident tables.
  __builtin_nontemporal_store(f, (v4f*)(out + (size_t)tid * 4u));
}

extern "C" void kernel_launch(void* const* d_in, const int* in_sizes, int n_in,
                              void* d_out, int out_size, void* d_ws, size_t ws_size,
                              hipStream_t stream) {
  (void)d_ws; (void)ws_size; (void)out_size; (void)n_in;

  const float* coords = (const float*)d_in[0];
  float* out = (float*)d_out;
  const unsigned int nPoints = (unsigned int)(in_sizes[0] / 4);

  // Mirror the reference's level-parameter computation in double precision,
  // but take the hashmap sizes from the actual table shapes (ground truth),
  // reconciling sr where the size is below the 2^20 clamp.
  Params p;
  const double g = std::exp((std::log(512.0) - std::log(8.0)) / 15.0);
  auto levelSize = [](long long s) -> long long {
    long long t = s < 32 ? s : 32;
    long long v = s * s * s * t;
    return v < (1LL << 20) ? v : (1LL << 20);
  };
  for (int l = 0; l < 16; ++l) {
    const double x = 8.0 * std::pow(g, (double)l);
    int sr = (int)std::floor(x);
    const unsigned int actual = (unsigned int)(in_sizes[1 + l] / 4);
    if ((long long)actual != levelSize(sr)) {
      for (int d = -2; d <= 2; ++d) {
        if (d == 0) continue;
        if (levelSize(sr + d) == (long long)actual) { sr += d; break; }
      }
    }
    const int tr = sr < 32 ? sr : 32;
    p.lv[l].M = 0xFFFFFFFFFFFFFFFFull / (unsigned long long)actual + 1ull;
    p.lv[l].tbl = (unsigned long long)d_in[1 + l];
    p.lv[l].sxyz = (float)sr;
    p.lv[l].st = (float)tr;
    p.lv[l].size = actual;
    p.lv[l]._pad = 0;
  }

  const unsigned int nPairs = nPoints * 16u;  // one thread per (point, level)
  dim3 block(256);
  dim3 grid((nPairs + 255u) / 256u);
  temporal_hash_encode<<<grid, block, 0, stream>>>(coords, out, p, nPoints);
}